// CapsuleLayer_66666482368714
// MI455X (gfx1250) — compile-verified
//
#include <hip/hip_runtime.h>
#include <math.h>

// CDNA5 / gfx1250 fused CapsuleLayer (votes einsum via V_WMMA_F32_16X16X4_F32 + EM routing in LDS)

typedef __attribute__((ext_vector_type(2))) float v2f;
typedef __attribute__((ext_vector_type(8))) float v8f;

#define NUM_IN  32
#define NUM_OUT 32
#define DD      16          // pose dim = 4x4
#define DIM_IN  17
#define HH      14
#define WW      14
#define BB      16
#define EPSF    1e-7f
#define TWO_PI_F 6.283185307179586f

// LDS layout (floats)
#define OFF_VOT   0                     // [32][32][16] votes, 16384 f
#define OFF_PL    (OFF_VOT + 16384)     // [32][16] pose_in, 512 f
#define OFF_CRAW  (OFF_PL  + 512)       // [32]
#define OFF_CN    (OFF_CRAW + 32)       // [32]
#define OFF_PRI   (OFF_CN  + 32)        // [32][32] priors, 1024 f
#define OFF_LPDF  (OFF_PRI + 1024)      // [32][32] log pdf, 1024 f
#define OFF_MU    (OFF_LPDF + 1024)     // [32][16]
#define OFF_SG    (OFF_MU  + 512)       // [32][16]
#define OFF_SP    (OFF_SG  + 512)       // [32] sum_priors
#define OFF_SPO   (OFF_SP  + 32)        // [32] sum_priors_out
#define LDS_FLOATS (OFF_SPO + 32)       // 20096 floats = 80384 bytes

__launch_bounds__(256)
__global__ void capsule_em_fused(const float* __restrict__ g_in,
                                 const float* __restrict__ g_W,
                                 const float* __restrict__ g_bias,
                                 float* __restrict__ g_out)
{
    extern __shared__ float smem[];
    float* vot  = smem + OFF_VOT;    // vot[m*512 + n*16 + d], d = i*4 + k
    float* Pl   = smem + OFF_PL;     // Pl[m*16 + j*4 + k]
    float* craw = smem + OFF_CRAW;
    float* cN   = smem + OFF_CN;
    float* pri  = smem + OFF_PRI;
    float* lpdf = smem + OFF_LPDF;
    float* mu_s = smem + OFF_MU;
    float* sg_s = smem + OFF_SG;
    float* sp   = smem + OFF_SP;
    float* spo  = smem + OFF_SPO;

    const int tid = threadIdx.x;
    const int pos = blockIdx.x;                 // b*196 + h*14 + w
    const int w   = pos % WW;
    const int h   = (pos / WW) % HH;

    const float* in_p = g_in + (size_t)pos * (NUM_IN * DIM_IN);
    const float* W_hw = g_W + (size_t)(h * WW + w) * (NUM_IN * NUM_OUT * 16);

    // ---- stage pose & raw coefficients into LDS ----
    // conv[m][dim] = inputs[dim*32 + m]; pose: dim = j*4+k (0..15), coeff: dim 16
    for (int idx = tid; idx < NUM_IN * 16; idx += 256) {
        int m = idx >> 4, e = idx & 15;
        Pl[idx] = in_p[e * NUM_IN + m];
    }
    if (tid < NUM_IN) craw[tid] = in_p[16 * NUM_IN + tid];
    __syncthreads();
    if (tid < NUM_IN) {             // normalize coeffs over m (redundant per lane, cheap)
        float s = 0.f;
        #pragma unroll
        for (int m = 0; m < NUM_IN; ++m) s += craw[m];
        cN[tid] = craw[tid] / s;
    }

    // ---- votes[m,n,i,k] = sum_j W[m,(n,i),j] * P[m,j,k]  via f32 WMMA 16x16x4 ----
    // Per m: A = W[m] rows (n*4+i) tiled 8x16 rows, K=j(4); B = P[m] (4x4 in a 4x16 tile).
    const int lane = tid & 31;
    const int wv   = tid >> 5;       // wave id 0..7 = row-tile index t
    const int colN = lane & 15;      // N column
    const int hlf  = lane >> 4;      // 0: K rows {0,1}; 1: K rows {2,3} (ISA 32-bit A/B layout)
    const int j0   = hlf * 2;

    for (int m = 0; m < NUM_IN; ++m) {
        const float* Wm = W_hw + m * (NUM_OUT * 16);   // 512 contiguous f: row stride 4, j innermost
        // prefetch next m's W slice (global_prefetch_b8)
        __builtin_prefetch(Wm + ((m + 1 < NUM_IN) ? NUM_OUT * 16 : 0), 0, 1);

        // A operand: lane holds row R = wv*16 + colN, elems j0, j0+1 (8B aligned)
        const int R = wv * 16 + colN;
        v2f a = *(const v2f*)(Wm + R * 4 + j0);

        // B operand: B[jrow][colN]; only cols 0..3 valid, rest zero
        int cc = (colN < 4) ? colN : 0;
        float bx = Pl[m * 16 + (j0    ) * 4 + cc];
        float by = Pl[m * 16 + (j0 + 1) * 4 + cc];
        v2f bmat;
        bmat.x = (colN < 4) ? bx : 0.f;
        bmat.y = (colN < 4) ? by : 0.f;

        v8f acc = {};
        acc = __builtin_amdgcn_wmma_f32_16x16x4_f32(false, a, false, bmat,
                                                    (short)0, acc, false, false);

        // D layout: VGPR r -> rows (r | r+8), col = lane; keep cols 0..3
        if (colN < 4) {
            float* vm = vot + m * (NUM_OUT * DD);
            #pragma unroll
            for (int r = 0; r < 8; ++r) {
                int row = wv * 16 + hlf * 8 + r;       // row = n*4 + i
                vm[row * 4 + colN] = acc[r];           // vot[m][n*16 + i*4 + k]
            }
        }
    }
    __syncthreads();

    // ---- EM routing, 3 iterations ----
    for (int it = 0; it < 3; ++it) {
        // priors = coeffs * priors  (initial priors = 1/num_out)
        for (int idx = tid; idx < NUM_IN * NUM_OUT; idx += 256) {
            int m = idx >> 5;
            float pv = (it == 0) ? (1.0f / NUM_OUT) : pri[idx];
            pri[idx] = pv * cN[m];
        }
        __syncthreads();
        // sum_priors over m
        if (tid < NUM_OUT) {
            float s = 0.f;
            #pragma unroll
            for (int m = 0; m < NUM_IN; ++m) s += pri[m * NUM_OUT + tid];
            sp[tid] = s;
        }
        __syncthreads();
        // priors_in = priors / sum_priors (overwrite; next-iter priors come from softmax)
        for (int idx = tid; idx < NUM_IN * NUM_OUT; idx += 256)
            pri[idx] = pri[idx] / sp[idx & 31];
        __syncthreads();
        // mu[n,d] = sum_m priors_in * votes
        for (int idx = tid; idx < NUM_OUT * DD; idx += 256) {
            int n = idx >> 4;
            float acc = 0.f;
            #pragma unroll
            for (int m = 0; m < NUM_IN; ++m)
                acc += pri[m * NUM_OUT + n] * vot[m * (NUM_OUT * DD) + idx];
            mu_s[idx] = acc;
        }
        __syncthreads();
        // sigma_sq[n,d] = sum_m priors_in * (v - mu)^2
        for (int idx = tid; idx < NUM_OUT * DD; idx += 256) {
            int n = idx >> 4;
            float mu = mu_s[idx];
            float acc = 0.f;
            #pragma unroll
            for (int m = 0; m < NUM_IN; ++m) {
                float dv = vot[m * (NUM_OUT * DD) + idx] - mu;
                acc += pri[m * NUM_OUT + n] * dv * dv;
            }
            sg_s[idx] = acc;
        }
        __syncthreads();
        if (it == 2) break;

        // log_pdf[m,n]
        for (int idx = tid; idx < NUM_IN * NUM_OUT; idx += 256) {
            int m = idx >> 5, n = idx & 31;
            float s = 0.f;
            #pragma unroll
            for (int d = 0; d < DD; ++d) {
                float dv = vot[m * (NUM_OUT * DD) + n * DD + d] - mu_s[n * DD + d];
                float sg = sg_s[n * DD + d];
                s += dv * dv / sg + __logf(sg * TWO_PI_F + EPSF);
            }
            lpdf[idx] = -0.5f * s;
        }
        // sum_priors_out = sum_priors / total
        if (tid < NUM_OUT) {
            float tot = 0.f;
            #pragma unroll
            for (int n = 0; n < NUM_OUT; ++n) tot += sp[n];
            spo[tid] = sp[tid] / tot;
        }
        __syncthreads();
        // priors = softmax_n( log(spo+eps) + log_pdf )  -- one thread per m
        if (tid < NUM_IN) {
            int m = tid;
            float mx = -3.4e38f;
            for (int n = 0; n < NUM_OUT; ++n) {
                float l = __logf(spo[n] + EPSF) + lpdf[m * NUM_OUT + n];
                mx = fmaxf(mx, l);
            }
            float s = 0.f;
            for (int n = 0; n < NUM_OUT; ++n) {
                float l = __logf(spo[n] + EPSF) + lpdf[m * NUM_OUT + n];
                float e = __expf(l - mx);
                pri[m * NUM_OUT + n] = e;
                s += e;
            }
            float inv = 1.0f / s;
            for (int n = 0; n < NUM_OUT; ++n) pri[m * NUM_OUT + n] *= inv;
        }
        __syncthreads();
    }

    // ---- activations + outputs ----
    float* outp = g_out + (size_t)pos * (NUM_OUT * DD + NUM_OUT);
    if (tid < NUM_OUT) {
        int n = tid;
        float s = 0.f;
        #pragma unroll
        for (int d = 0; d < DD; ++d)
            s += 1.0f + __logf(sg_s[n * DD + d] * TWO_PI_F + EPSF);
        float x = g_bias[n] * (float)DD - 0.5f * spo[n] * s;   // spo from iter 1 (last computed)
        outp[NUM_OUT * DD + n] = 1.0f / (1.0f + __expf(-x));
    }
    for (int idx = tid; idx < NUM_OUT * DD; idx += 256)
        outp[idx] = mu_s[idx];                                  // poses: n slow, d fast
}

extern "C" void kernel_launch(void* const* d_in, const int* in_sizes, int n_in,
                              void* d_out, int out_size, void* d_ws, size_t ws_size,
                              hipStream_t stream) {
    const float* inputs = (const float*)d_in[0];   // (16,14,14,544) f32
    const float* Wt     = (const float*)d_in[1];   // (14,14,32,32,4,4) f32
    const float* bias   = (const float*)d_in[2];   // (1,32,1) f32
    float* out = (float*)d_out;                    // (16,14,14,544) f32

    dim3 grid(BB * HH * WW);                       // 3136 workgroups, one per (b,h,w)
    size_t lds_bytes = (size_t)LDS_FLOATS * sizeof(float);   // ~80 KB (<< 320 KB/WGP)
    capsule_em_fused<<<grid, 256, lds_bytes, stream>>>(inputs, Wt, bias, out);
}